// NLMConv1DBlockV2_15857019257194
// MI455X (gfx1250) — compile-verified
//
#include <hip/hip_runtime.h>
#include <hip/hip_bf16.h>
#include <math.h>

typedef float v2f __attribute__((ext_vector_type(2)));
typedef float v8f __attribute__((ext_vector_type(8)));

#define B_   2
#define T_   16
#define N_   128
#define CIN  64
#define OUT_ 128
#define KSTEPS (CIN / 4)   // 16 WMMA k-steps per 64-wide chunk

// ---------------------------------------------------------------------------
// GEMM for a = xc@W1 and bpart = xc@W2.
// Rows r = (b*T + t)*N + n (4096). K=192 as 3 chunks of 64 over x; chunk0
// skipped when t==0, chunk2 when t==T-1 (scalar branches via readfirstlane).
// One wave computes a 16x16 tile with V_WMMA_F32_16X16X4_F32 (full fp32).
// A frags preloaded once; B frags batch-loaded per chunk, then a dense
// 16-deep WMMA chain (one wait per chunk instead of one per WMMA).
// ---------------------------------------------------------------------------
__global__ __launch_bounds__(256) void gemm_ab_kernel(const float* __restrict__ x,
                                                      const float* __restrict__ W,
                                                      float* __restrict__ a_buf,
                                                      float* __restrict__ b_buf) {
  const int lane = threadIdx.x & 31;
  // Wave-uniform task id forced into SGPRs.
  const int task = __builtin_amdgcn_readfirstlane(blockIdx.x * 8 + (threadIdx.x >> 5));
  const int rt = task >> 4;                 // 256 row tiles
  const int ct = task & 15;                 // 16 col tiles
  const int r0 = rt * 16;
  const int t  = (r0 / N_) % T_;

  float* __restrict__ dst = (ct < 8) ? a_buf : b_buf;
  const int cbase = (ct < 8) ? 0 : 3 * CIN;       // W1 rows vs W2 rows
  const int obase = (ct & 7) * 16;

  const int mrow  = lane & 15;                    // A: M index / B: N index
  const int khalf = lane >> 4;                    // 0 -> K{0,1}, 1 -> K{2,3}

  // Preload the full A slab for this tile: A[M][k0+2*khalf .. +1], k0=0,4,...,60
  const float* __restrict__ aptr = x + (r0 + mrow) * CIN + khalf * 2;
  v2f afr[KSTEPS];
#pragma unroll
  for (int k = 0; k < KSTEPS; ++k) afr[k] = *(const v2f*)(aptr + 4 * k);

  v8f acc = {0.f, 0.f, 0.f, 0.f, 0.f, 0.f, 0.f, 0.f};
#pragma unroll
  for (int q = 0; q < 3; ++q) {
    if (q == 0 && t == 0) continue;          // scalar (SCC) branch
    if (q == 2 && t == T_ - 1) continue;     // scalar (SCC) branch
    const float* __restrict__ wbase = W + (cbase + q * CIN + khalf * 2) * OUT_ + obase + mrow;
    v2f bfr[KSTEPS];
#pragma unroll
    for (int k = 0; k < KSTEPS; ++k) {       // one clause of loads
      bfr[k].x = wbase[4 * k * OUT_];        // B[k'][n]
      bfr[k].y = wbase[(4 * k + 1) * OUT_];  // B[k'+1][n]
    }
#pragma unroll
    for (int k = 0; k < KSTEPS; ++k) {       // dense WMMA chain
      acc = __builtin_amdgcn_wmma_f32_16x16x4_f32(false, afr[k], false, bfr[k],
                                                  (short)0, acc, false, false);
    }
  }
#pragma unroll
  for (int v = 0; v < 8; ++v) {
    const int M = v + khalf * 8;             // C/D layout: M = v + 8*(lane>=16)
    dst[(r0 + M) * OUT_ + obase + mrow] = acc[v];
  }
}

// ---------------------------------------------------------------------------
// cpart[b,i,j,:] = cond[b,i,j,:]@Wc1 + cond[b,j,i,:]@Wc2
// Rows r = (b*N + i)*N + j (32768). Two K=64 terms into one accumulator,
// same batched-load + dense-WMMA-chain structure.
// ---------------------------------------------------------------------------
__global__ __launch_bounds__(256) void gemm_cpart_kernel(const float* __restrict__ cond,
                                                         const float* __restrict__ W,
                                                         float* __restrict__ cp_buf) {
  const int lane = threadIdx.x & 31;
  const int task = __builtin_amdgcn_readfirstlane(blockIdx.x * 8 + (threadIdx.x >> 5));
  const int rt = task >> 3;                 // 2048 row tiles
  const int ct = task & 7;                  // 8 col tiles
  const int r0 = rt * 16;
  const int bb  = r0 / (N_ * N_);
  const int rem = r0 - bb * N_ * N_;
  const int i   = rem / N_;
  const int j0  = rem - i * N_;
  const int obase = ct * 16;

  const int mrow  = lane & 15;
  const int khalf = lane >> 4;

  const float* __restrict__ a1 = cond + (r0 + mrow) * CIN + khalf * 2;                      // cond[b,i,j0+M,:]
  const float* __restrict__ a2 = cond + ((bb * N_ + j0 + mrow) * N_ + i) * CIN + khalf * 2; // cond[b,j0+M,i,:]

  v2f af1[KSTEPS], af2[KSTEPS];
#pragma unroll
  for (int k = 0; k < KSTEPS; ++k) {
    af1[k] = *(const v2f*)(a1 + 4 * k);
    af2[k] = *(const v2f*)(a2 + 4 * k);
  }

  v8f acc = {0.f, 0.f, 0.f, 0.f, 0.f, 0.f, 0.f, 0.f};
#pragma unroll
  for (int term = 0; term < 2; ++term) {
    const float* __restrict__ wbase =
        W + ((6 + term) * CIN + khalf * 2) * OUT_ + obase + mrow;   // Wc1 / Wc2
    v2f bfr[KSTEPS];
#pragma unroll
    for (int k = 0; k < KSTEPS; ++k) {
      bfr[k].x = wbase[4 * k * OUT_];
      bfr[k].y = wbase[(4 * k + 1) * OUT_];
    }
#pragma unroll
    for (int k = 0; k < KSTEPS; ++k) {
      const v2f a = (term == 0) ? af1[k] : af2[k];
      acc = __builtin_amdgcn_wmma_f32_16x16x4_f32(false, a, false, bfr[k],
                                                  (short)0, acc, false, false);
    }
  }
#pragma unroll
  for (int v = 0; v < 8; ++v) {
    const int M = v + khalf * 8;
    cp_buf[(r0 + M) * OUT_ + obase + mrow] = acc[v];
  }
}

// ---------------------------------------------------------------------------
// Fused epilogue: out[b,t,i,o] = max_j silu(a + bpart[j] + cpart[i,j] + bias)
// Block = one (b,t,i). 128 threads = 4 j-slices x 32 lanes; each lane owns 4
// contiguous o via float4 loads; j unrolled x2; LDS max-reduce across slices.
// ---------------------------------------------------------------------------
__global__ __launch_bounds__(128) void fused_silu_max_kernel(const float* __restrict__ a_buf,
                                                             const float* __restrict__ b_buf,
                                                             const float* __restrict__ cp_buf,
                                                             const float* __restrict__ bias,
                                                             float* __restrict__ out) {
  const int blk  = blockIdx.x;            // (b*T + t)*N + i,  4096 blocks
  const int i    = blk % N_;
  const int bt   = blk / N_;              // b*T + t
  const int bb   = bt / T_;
  const int lane = threadIdx.x & 31;
  const int sl   = threadIdx.x >> 5;      // j-slice 0..3
  const int o4   = lane * 4;

  float4 av = ((const float4*)(a_buf + (size_t)blk * OUT_))[lane];
  float4 bv = ((const float4*)bias)[lane];
  av.x += bv.x; av.y += bv.y; av.z += bv.z; av.w += bv.w;

  const float* __restrict__ bp_base = b_buf + (size_t)bt * N_ * OUT_;
  const float* __restrict__ cp_base = cp_buf + ((size_t)(bb * N_ + i)) * N_ * OUT_;

  float4 m = {-INFINITY, -INFINITY, -INFINITY, -INFINITY};
#pragma unroll 2
  for (int j = sl; j < N_; j += 4) {
    if (j + 8 < N_) {
      __builtin_prefetch((const void*)(cp_base + (j + 8) * OUT_ + o4), 0, 1);
    }
    float4 bp = *(const float4*)(bp_base + j * OUT_ + o4);
    float4 cp = *(const float4*)(cp_base + j * OUT_ + o4);
    float s0 = av.x + bp.x + cp.x;
    float s1 = av.y + bp.y + cp.y;
    float s2 = av.z + bp.z + cp.z;
    float s3 = av.w + bp.w + cp.w;
    m.x = fmaxf(m.x, s0 / (1.0f + __expf(-s0)));
    m.y = fmaxf(m.y, s1 / (1.0f + __expf(-s1)));
    m.z = fmaxf(m.z, s2 / (1.0f + __expf(-s2)));
    m.w = fmaxf(m.w, s3 / (1.0f + __expf(-s3)));
  }

  __shared__ float4 red[4][32];
  red[sl][lane] = m;
  __syncthreads();
  if (sl == 0) {
    float4 r0 = red[0][lane], r1 = red[1][lane], r2 = red[2][lane], r3 = red[3][lane];
    float4 o;
    o.x = fmaxf(fmaxf(r0.x, r1.x), fmaxf(r2.x, r3.x));
    o.y = fmaxf(fmaxf(r0.y, r1.y), fmaxf(r2.y, r3.y));
    o.z = fmaxf(fmaxf(r0.z, r1.z), fmaxf(r2.z, r3.z));
    o.w = fmaxf(fmaxf(r0.w, r1.w), fmaxf(r2.w, r3.w));
    ((float4*)(out + (size_t)blk * OUT_))[lane] = o;
  }
}

extern "C" void kernel_launch(void* const* d_in, const int* in_sizes, int n_in,
                              void* d_out, int out_size, void* d_ws, size_t ws_size,
                              hipStream_t stream) {
  const float* x    = (const float*)d_in[0];   // (2,16,128,64)
  const float* cond = (const float*)d_in[1];   // (2,128,128,64)
  const float* W    = (const float*)d_in[2];   // (512,128)
  const float* bias = (const float*)d_in[3];   // (128,)
  float* out = (float*)d_out;                  // (2,16,128,128)

  float* a_buf  = (float*)d_ws;                                  // 4096*128
  float* b_buf  = a_buf + (size_t)B_ * T_ * N_ * OUT_;           // 4096*128
  float* cp_buf = b_buf + (size_t)B_ * T_ * N_ * OUT_;           // 32768*128

  // 4096 wave-tiles / 8 waves per block
  gemm_ab_kernel<<<512, 256, 0, stream>>>(x, W, a_buf, b_buf);
  // 16384 wave-tiles / 8 waves per block
  gemm_cpart_kernel<<<2048, 256, 0, stream>>>(cond, W, cp_buf);
  // one block per (b,t,i)
  fused_silu_max_kernel<<<B_ * T_ * N_, 128, 0, stream>>>(a_buf, b_buf, cp_buf, bias, out);
}